// SSL_GNN_38130719654217
// MI455X (gfx1250) — compile-verified
//
#include <hip/hip_runtime.h>
#include <hip/hip_bf16.h>

typedef __attribute__((ext_vector_type(2))) float v2f;
typedef __attribute__((ext_vector_type(8))) float v8f;
typedef __attribute__((address_space(3))) char lds_byte;

#define DF 128

// ---------------------------------------------------------------------------
// GEMM: C[M x 128] = A[M x 128] * W[128 x 128] (+bias) (+relu)
// f32 WMMA 16x16x4. Block = 256 threads = 8 waves; each wave computes a
// 16-row x 128-col output strip (8 col tiles, 64 accumulator VGPRs).
// W is staged in LDS (64 KB) once per block via async-to-LDS DMA.
// ---------------------------------------------------------------------------
__global__ __launch_bounds__(256) void gemm128_wmma(
    const float* __restrict__ A, const float* __restrict__ W,
    float* __restrict__ C, const float* __restrict__ bias,
    int relu, int M)
{
    __shared__ float Wl[DF * DF];
    // Async staging: each thread issues 16 x b128 global->LDS transfers
    // (ASYNCcnt-tracked, no VGPR round trip). The AS(3) pointer operand keeps
    // pointer provenance so the compiler knows the asm writes Wl.
    {
        const float4* Wv  = (const float4*)W;
        float4*       Wlv = (float4*)Wl;
        for (int i = threadIdx.x; i < DF * DF / 4; i += 256) {
            lds_byte* ldst = (lds_byte*)&Wlv[i];   // addrspacecast: value = LDS byte offset
            unsigned long long g = (unsigned long long)(uintptr_t)&Wv[i];
            asm volatile("global_load_async_to_lds_b128 %0, %1, off"
                         :: "v"(ldst), "v"(g) : "memory");
        }
        asm volatile("s_wait_asynccnt 0x0" ::: "memory");
    }
    __syncthreads();

    const int wave = threadIdx.x >> 5;
    const int lane = threadIdx.x & 31;
    const int tile = blockIdx.x * 8 + wave;     // 16-row tile index
    const int numTiles = M >> 4;                // M divisible by 16 here
    if (tile >= numTiles) return;               // wave-uniform exit (EXEC stays full)

    const int row0 = tile << 4;
    const int m    = lane & 15;                 // A: lane -> row within tile
    const int kg   = (lane >> 4) << 1;          // lane half selects K pair (0 or 2)

    v8f acc[8] = {};

    const float* Arow = A + (size_t)(row0 + m) * DF + kg;

    // Software-pipelined A fragments: issue next iteration's global load a full
    // 8-WMMA batch ahead of its use.
    v2f a = *(const v2f*)(Arow);
    for (int k0 = 0; k0 < DF; k0 += 4) {
        const int kn = (k0 + 4 < DF) ? (k0 + 4) : 0;   // uniform; safe dummy on last iter
        v2f anext = *(const v2f*)(Arow + kn);
        const int kk = k0 + kg;
        #pragma unroll
        for (int t = 0; t < 8; ++t) {
            const int n = (lane & 15) + (t << 4);
            // B fragment 4x16: lane n holds {W[kk][n], W[kk+1][n]}
            v2f b;
            b.x = Wl[kk * DF + n];
            b.y = Wl[(kk + 1) * DF + n];
            acc[t] = __builtin_amdgcn_wmma_f32_16x16x4_f32(
                false, a, false, b, (short)0, acc[t], false, false);
        }
        a = anext;
    }

    // C layout: VGPR v -> row v (lanes 0-15) / row v+8 (lanes 16-31), col = lane&15
    const int colBase = lane & 15;
    const int rowOff  = (lane >> 4) << 3;
    #pragma unroll
    for (int t = 0; t < 8; ++t) {
        const int col = colBase + (t << 4);
        const float bv = bias ? bias[col] : 0.0f;
        #pragma unroll
        for (int v = 0; v < 8; ++v) {
            float val = acc[t][v] + bv;
            if (relu) val = fmaxf(val, 0.0f);
            C[(size_t)(row0 + rowOff + v) * DF + col] = val;
        }
    }
}

// ---------------------------------------------------------------------------
// Degree / normalization kernels
// ---------------------------------------------------------------------------
__global__ void deg_init_k(float* deg, int n) {
    int i = blockIdx.x * blockDim.x + threadIdx.x;
    if (i < n) deg[i] = 1.0f;                   // self-loop
}
__global__ void deg_count_k(const int* __restrict__ dst, float* deg, int E) {
    int i = blockIdx.x * blockDim.x + threadIdx.x;
    if (i < E) atomicAdd(&deg[dst[i]], 1.0f);
}
__global__ void deg_rsqrt_k(float* deg, int n) {
    int i = blockIdx.x * blockDim.x + threadIdx.x;
    if (i < n) deg[i] = rsqrtf(deg[i]);         // deg -> D^{-1/2} in place
}

// agg[i,:] = h[i,:] * dis[i]^2   (self-loop contribution, also zero-fills agg)
__global__ void self_init_k(const float* __restrict__ h, const float* __restrict__ dis,
                            float* __restrict__ agg, int n) {
    int i = blockIdx.x * blockDim.x + threadIdx.x;
    if (i < n * DF) {
        float d = dis[i >> 7];
        agg[i] = h[i] * d * d;
    }
}

// One wave (32 lanes) per edge: each lane moves 4 features (float4 gather,
// 4x global_atomic_add_f32 scatter).
__global__ __launch_bounds__(256) void edge_agg_k(
    const float* __restrict__ h, const float* __restrict__ dis,
    const int* __restrict__ src, const int* __restrict__ dst,
    float* __restrict__ agg, int E)
{
    const int wave = threadIdx.x >> 5;
    const int lane = threadIdx.x & 31;
    const int e = blockIdx.x * 8 + wave;
    if (e >= E) return;
    const int s = src[e], d = dst[e];
    const float norm = dis[s] * dis[d];
    float4 v = ((const float4*)(h + (size_t)s * DF))[lane];
    float* ad = agg + (size_t)d * DF + lane * 4;
    atomicAdd(ad + 0, v.x * norm);
    atomicAdd(ad + 1, v.y * norm);
    atomicAdd(ad + 2, v.z * norm);
    atomicAdd(ad + 3, v.w * norm);
}

__global__ void bias_relu_k(float* __restrict__ x, const float* __restrict__ b, int total) {
    int i = blockIdx.x * blockDim.x + threadIdx.x;
    if (i < total) x[i] = fmaxf(x[i] + b[i & (DF - 1)], 0.0f);
}

// ---------------------------------------------------------------------------
// One full GCN forward pass
// ---------------------------------------------------------------------------
static void run_forward(const float* x, const int* ei,
                        const float* W_in, const float* b_in,
                        const float* W_h,  const float* b_h,
                        const float* W_out,const float* b_out,
                        const float* Wf1,  const float* bf1,
                        const float* Wf2,  const float* bf2,
                        float* out, float* bufA, float* bufB, float* dis,
                        int N, int E, hipStream_t s)
{
    const int tN  = (N + 255) / 256;
    const int tE  = (E + 255) / 256;
    const int tNF = (N * DF + 255) / 256;
    const int gemmBlocks = ((N >> 4) + 7) / 8;
    const int edgeBlocks = (E + 7) / 8;

    deg_init_k <<<tN, 256, 0, s>>>(dis, N);
    deg_count_k<<<tE, 256, 0, s>>>(ei + E, dis, E);   // dst row of edge_index
    deg_rsqrt_k<<<tN, 256, 0, s>>>(dis, N);

    const float* Ws[3] = {W_in, W_h, W_out};
    const float* bs[3] = {b_in, b_h, b_out};
    const float* cur = x;
    for (int l = 0; l < 3; ++l) {
        gemm128_wmma<<<gemmBlocks, 256, 0, s>>>(cur, Ws[l], bufA, nullptr, 0, N); // h = cur @ W
        self_init_k <<<tNF, 256, 0, s>>>(bufA, dis, bufB, N);                      // self loop
        edge_agg_k  <<<edgeBlocks, 256, 0, s>>>(bufA, dis, ei, ei + E, bufB, E);   // scatter-add
        bias_relu_k <<<tNF, 256, 0, s>>>(bufB, bs[l], N * DF);                     // +b, relu
        cur = bufB;
    }
    gemm128_wmma<<<gemmBlocks, 256, 0, s>>>(cur,  Wf1, bufA, bf1, 1, N);  // relu(x@Wf1+bf1)
    gemm128_wmma<<<gemmBlocks, 256, 0, s>>>(bufA, Wf2, out,  bf2, 0, N);  // @Wf2 + bf2
}

extern "C" void kernel_launch(void* const* d_in, const int* in_sizes, int n_in,
                              void* d_out, int out_size, void* d_ws, size_t ws_size,
                              hipStream_t stream)
{
    const float* x      = (const float*)d_in[0];
    const int*   ei     = (const int*)  d_in[1];
    const float* x_aug  = (const float*)d_in[2];
    const int*   ei_aug = (const int*)  d_in[3];
    const float* W_in   = (const float*)d_in[4];
    const float* b_in   = (const float*)d_in[5];
    const float* W_h    = (const float*)d_in[6];
    const float* b_h    = (const float*)d_in[7];
    const float* W_out  = (const float*)d_in[8];
    const float* b_out  = (const float*)d_in[9];
    const float* Wf1    = (const float*)d_in[10];
    const float* bf1    = (const float*)d_in[11];
    const float* Wf2    = (const float*)d_in[12];
    const float* bf2    = (const float*)d_in[13];

    const int N = in_sizes[0] / DF;
    const int E = in_sizes[1] / 2;

    float* bufA = (float*)d_ws;                 // [N*128]
    float* bufB = bufA + (size_t)N * DF;        // [N*128]
    float* dis  = bufB + (size_t)N * DF;        // [N]

    float* out0 = (float*)d_out;
    float* out1 = out0 + (size_t)N * DF;

    run_forward(x,     ei,     W_in, b_in, W_h, b_h, W_out, b_out,
                Wf1, bf1, Wf2, bf2, out0, bufA, bufB, dis, N, E, stream);
    run_forward(x_aug, ei_aug, W_in, b_in, W_h, b_h, W_out, b_out,
                Wf1, bf1, Wf2, bf2, out1, bufA, bufB, dis, N, E, stream);
}